// CGSL_56487409877018
// MI455X (gfx1250) — compile-verified
//
#include <hip/hip_runtime.h>
#include <hip/hip_bf16.h>
#include <math.h>

#define B_    256
#define N_    200
#define NN_   (N_*N_)
#define FIN_  200
#define OUTF_ 256
#define NCLS_ 8
#define K_    3980

typedef float v2f __attribute__((ext_vector_type(2)));
typedef float v8f __attribute__((ext_vector_type(8)));

__device__ __forceinline__ unsigned ordkey(float f) {
  unsigned u = __float_as_uint(f);
  return (u & 0x80000000u) ? ~u : (u | 0x80000000u);
}

__device__ __forceinline__ v8f wmma4(v2f a, v2f b, v8f c) {
  return __builtin_amdgcn_wmma_f32_16x16x4_f32(false, a, false, b, (short)0, c,
                                               false, false);
}

// ---------------- Kernel A: gumbel + symmetrize -> ordered uint keys -------
__global__ void k_keys(const float* __restrict__ nets, const float* __restrict__ gu,
                       const int* __restrict__ net_index, unsigned* __restrict__ keys) {
  int idx = blockIdx.x * blockDim.x + threadIdx.x;
  if (idx >= B_ * NN_) return;
  int b = idx / NN_;
  int r = idx - b * NN_;
  int i = r / N_;
  int j = r - i * N_;
  if (i > j) return;  // only upper triangle (incl diag) participates in top-k
  const float* l  = nets + (size_t)net_index[b] * NN_;
  const float* ub = gu + (size_t)b * NN_;
  float u0 = ub[i * N_ + j], u1 = ub[j * N_ + i];
  u0 = fminf(fmaxf(u0, 1e-10f), 1.0f - 1e-10f);
  u1 = fminf(fmaxf(u1, 1e-10f), 1.0f - 1e-10f);
  float g0 = -logf(-logf(u0));
  float g1 = -logf(-logf(u1));
  // tau == 1.0
  float sym = 0.5f * ((l[i * N_ + j] + g0) + (l[j * N_ + i] + g1));
  keys[(size_t)b * NN_ + r] = ordkey(sym);
}

// ---------------- Kernel B: exact top-K via 4-pass radix select ------------
__global__ void k_select(const unsigned* __restrict__ keys, float* __restrict__ adj) {
  int b = blockIdx.x;
  const unsigned* kb = keys + (size_t)b * NN_;
  __shared__ unsigned hist[256];
  __shared__ unsigned s_pref;
  __shared__ int s_rem;
  int tid = threadIdx.x;

  unsigned prefix = 0;
  int remaining = K_;
  for (int shift = 24; shift >= 0; shift -= 8) {
    hist[tid] = 0;
    __syncthreads();
    for (int idx = tid; idx < NN_; idx += blockDim.x) {
      int i = idx / N_, j = idx - i * N_;
      if (i > j) continue;
      unsigned k = kb[idx];
      bool match = (shift == 24) || ((k >> (shift + 8)) == prefix);
      if (match) atomicAdd(&hist[(k >> shift) & 0xFFu], 1u);
    }
    __syncthreads();
    if (tid == 0) {
      int acc = 0;
      int v = 255;
      for (; v > 0; --v) {
        int c = (int)hist[v];
        if (acc + c >= remaining) break;
        acc += c;
      }
      s_pref = (prefix << 8) | (unsigned)v;
      s_rem = remaining - acc;
    }
    __syncthreads();
    prefix = s_pref;
    remaining = s_rem;
    __syncthreads();
  }

  unsigned tkey = prefix;          // K-th largest key
  int need_eq = remaining;         // how many entries equal to tkey to keep
  float* ab = adj + (size_t)b * NN_;
  for (int idx = tid; idx < NN_; idx += blockDim.x) {
    int i = idx / N_, j = idx - i * N_;
    if (i > j) continue;
    unsigned k = kb[idx];
    float sel = 0.0f;
    if (k > tkey) {
      sel = 1.0f;
    } else if (k == tkey) {
      // tie-break by smaller flat index, matching lax.top_k (ties are rare)
      int rank = 0;
      for (int q = 0; q < idx; ++q) {
        int qi = q / N_, qj = q - qi * N_;
        if (qi <= qj && kb[q] == tkey) rank++;
      }
      if (rank < need_eq) sel = 1.0f;
    }
    ab[i * N_ + j] = sel;   // symmetrized hard adjacency (diag written once)
    ab[j * N_ + i] = sel;
  }
}

// ---------------- Kernel C: a1 = adj @ x  (per-batch 200x200 @ 200x200) ----
// Wave tile: 32(M) x 32(N): 2 A fragments x 2 B fragments -> 4 WMMAs/K-step.
// Loads are address-clamped (never predicated); stores are guarded.
__global__ void k_gemm_ax(const float* __restrict__ adj, const float* __restrict__ x,
                          float* __restrict__ a1) {
  const int MT = 7, NC = 7;  // 7 M-tiles of 32, 7 N-chunks of 32 (224 >= 200)
  int wave = blockIdx.x * (blockDim.x >> 5) + (threadIdx.x >> 5);
  if (wave >= B_ * MT * NC) return;
  int b  = wave / (MT * NC);
  int rr = wave - b * MT * NC;
  int mt = rr / NC, nc = rr - mt * NC;
  int lane = threadIdx.x & 31;
  int half = lane >> 4, lm = lane & 15;
  int m0 = mt * 32 + lm;
  int m1 = m0 + 16;
  int m0l = m0 < N_ ? m0 : (N_ - 1);        // clamped A rows for loads
  int m1l = m1 < N_ ? m1 : (N_ - 1);
  int n0 = nc * 32 + lm;
  int n1 = n0 + 16;
  int n0l = n0 < N_ ? n0 : (N_ - 1);        // clamped B cols for loads
  int n1l = n1 < N_ ? n1 : (N_ - 1);
  const float* A0 = adj + (size_t)b * NN_ + (size_t)m0l * N_;
  const float* A1 = adj + (size_t)b * NN_ + (size_t)m1l * N_;
  const float* X  = x + (size_t)b * NN_;
  v8f c00 = {0.f,0.f,0.f,0.f,0.f,0.f,0.f,0.f};
  v8f c01 = c00, c10 = c00, c11 = c00;
#pragma unroll 2
  for (int k = 0; k < N_; k += 4) {         // K = 200, exact
    int ka = k + 2 * half;
    v2f a0 = *(const v2f*)(A0 + ka);        // 8B-aligned vector loads
    v2f a1f = *(const v2f*)(A1 + ka);
    const float* Xr = X + (size_t)ka * N_;
    v2f b0, b1;
    b0.x = Xr[n0l];        b1.x = Xr[n1l];
    b0.y = Xr[N_ + n0l];   b1.y = Xr[N_ + n1l];
    c00 = wmma4(a0,  b0, c00);
    c01 = wmma4(a0,  b1, c01);
    c10 = wmma4(a1f, b0, c10);
    c11 = wmma4(a1f, b1, c11);
  }
  float* C = a1 + (size_t)b * NN_;
  bool nv0 = n0 < N_, nv1 = n1 < N_;
  for (int r = 0; r < 8; ++r) {
    int mmA = mt * 32 + r + 8 * half;
    int mmB = mmA + 16;
    if (mmA < N_) {
      float* row = C + (size_t)mmA * N_;
      if (nv0) row[n0] = c00[r];
      if (nv1) row[n1] = c01[r];
    }
    if (mmB < N_) {
      float* row = C + (size_t)mmB * N_;
      if (nv0) row[n0] = c10[r];
      if (nv1) row[n1] = c11[r];
    }
  }
}

// ---------------- Kernel D: emb = relu(a1 @ W_gnn), M=51200,K=200,N=256 ----
// Wave tile: 32 x 32; all dims exact multiples -> zero guards anywhere.
__global__ void k_gemm_aw(const float* __restrict__ a1, const float* __restrict__ w,
                          float* __restrict__ emb) {
  const int MTILES = (B_ * N_) / 32;  // 1600
  const int NC = OUTF_ / 32;          // 8
  int wave = blockIdx.x * (blockDim.x >> 5) + (threadIdx.x >> 5);
  if (wave >= MTILES * NC) return;
  int mt = wave / NC, nc = wave - mt * NC;
  int lane = threadIdx.x & 31;
  int half = lane >> 4, lm = lane & 15;
  int m0 = mt * 32 + lm;
  int m1 = m0 + 16;
  int n0 = nc * 32 + lm;
  int n1 = n0 + 16;
  const float* A0 = a1 + (size_t)m0 * N_;
  const float* A1 = a1 + (size_t)m1 * N_;
  v8f c00 = {0.f,0.f,0.f,0.f,0.f,0.f,0.f,0.f};
  v8f c01 = c00, c10 = c00, c11 = c00;
#pragma unroll 2
  for (int k = 0; k < FIN_; k += 4) {  // K = 200
    int ka = k + 2 * half;
    v2f a0 = *(const v2f*)(A0 + ka);
    v2f a1f = *(const v2f*)(A1 + ka);
    const float* Wr = w + (size_t)ka * OUTF_;
    v2f b0, b1;
    b0.x = Wr[n0];          b1.x = Wr[n1];
    b0.y = Wr[OUTF_ + n0];  b1.y = Wr[OUTF_ + n1];
    c00 = wmma4(a0,  b0, c00);
    c01 = wmma4(a0,  b1, c01);
    c10 = wmma4(a1f, b0, c10);
    c11 = wmma4(a1f, b1, c11);
  }
  for (int r = 0; r < 8; ++r) {
    int mmA = mt * 32 + r + 8 * half;
    int mmB = mmA + 16;
    float* rowA = emb + (size_t)mmA * OUTF_;
    float* rowB = emb + (size_t)mmB * OUTF_;
    float v;
    v = c00[r]; rowA[n0] = v > 0.f ? v : 0.f;
    v = c01[r]; rowA[n1] = v > 0.f ? v : 0.f;
    v = c10[r]; rowB[n0] = v > 0.f ? v : 0.f;
    v = c11[r]; rowB[n1] = v > 0.f ? v : 0.f;
  }
}

// ---------------- Kernel E: out = emb @ w_lin + b, M=51200,K=256,N=8(pad16)
__global__ void k_gemm_out(const float* __restrict__ emb, const float* __restrict__ wl,
                           const float* __restrict__ bl, float* __restrict__ out) {
  const int MTILES = (B_ * N_) / 16;  // 3200
  int wave = blockIdx.x * (blockDim.x >> 5) + (threadIdx.x >> 5);
  if (wave >= MTILES) return;
  int mt = wave;
  int lane = threadIdx.x & 31;
  int half = lane >> 4, lm = lane & 15;
  int m  = mt * 16 + lm;
  int n  = lm;                          // only n<8 are real columns
  int nl = n < NCLS_ ? n : (NCLS_ - 1); // clamped load column
  const float* A = emb + (size_t)m * OUTF_;
  v8f c = {0.f,0.f,0.f,0.f,0.f,0.f,0.f,0.f};
#pragma unroll 2
  for (int k = 0; k < OUTF_; k += 4) {  // K = 256
    int ka = k + 2 * half;
    v2f a = *(const v2f*)(A + ka);
    v2f bm;
    bm.x = wl[ka * NCLS_ + nl];
    bm.y = wl[(ka + 1) * NCLS_ + nl];
    c = wmma4(a, bm, c);
  }
  if (n < NCLS_) {
    float bias = bl[n];
    for (int r = 0; r < 8; ++r) {
      int mm = mt * 16 + r + 8 * half;
      out[(size_t)mm * NCLS_ + n] = c[r] + bias;
    }
  }
}

extern "C" void kernel_launch(void* const* d_in, const int* in_sizes, int n_in,
                              void* d_out, int out_size, void* d_ws, size_t ws_size,
                              hipStream_t stream) {
  const float* x      = (const float*)d_in[0];
  const float* gu     = (const float*)d_in[1];
  const float* nets   = (const float*)d_in[2];
  const float* w_gnn  = (const float*)d_in[3];
  const float* w_lin  = (const float*)d_in[4];
  const float* b_lin  = (const float*)d_in[5];
  const int*   net_ix = (const int*)d_in[6];

  float* out = (float*)d_out;                              // [B,N,8]
  float* emb = out + (size_t)B_ * N_ * NCLS_;              // [B,N,256]
  float* adj = emb + (size_t)B_ * N_ * OUTF_;              // [B,N,N]

  // keys temporarily live in the emb output region (52.4 MB >= 41 MB);
  // emb is rewritten by k_gemm_aw after the keys are consumed.
  unsigned* keys = (unsigned*)emb;
  float* a1 = (float*)d_ws;                                // [B,N,N] = 41 MB

  // A: keys (B*N*N threads, upper-tri active)
  k_keys<<<(B_ * NN_) / 256, 256, 0, stream>>>(nets, gu, net_ix, keys);
  // B: per-batch exact top-K radix select -> hard symmetric adjacency
  k_select<<<B_, 256, 0, stream>>>(keys, adj);
  // C: a1 = adj @ x   (256*7*7 = 12544 waves, 8 waves/block -> 1568 blocks)
  k_gemm_ax<<<1568, 256, 0, stream>>>(adj, x, a1);
  // D: emb = relu(a1 @ W_gnn)  (1600*8 = 12800 waves -> 1600 blocks)
  k_gemm_aw<<<1600, 256, 0, stream>>>(a1, w_gnn, emb);
  // E: out = emb @ w_lin + b   (3200 waves -> 400 blocks)
  k_gemm_out<<<400, 256, 0, stream>>>(emb, w_lin, b_lin, out);
}